// MultiHeadMHC_13950053777920
// MI455X (gfx1250) — compile-verified
//
#include <hip/hip_runtime.h>

typedef float v2f __attribute__((ext_vector_type(2)));
typedef float v8f __attribute__((ext_vector_type(8)));

#define KEPS 1e-8f
#define SINKHORN_ITERS 50

__global__ __launch_bounds__(256) void mhc_sinkhorn_mix_kernel(
    const float* __restrict__ states,   // [B, 16, 1024]
    const float* __restrict__ attn,     // [B, 16, 16]
    float* __restrict__ out)            // [B, 1024]
{
    const int b    = blockIdx.x;
    const int tid  = threadIdx.x;
    const int lane = tid & 31;
    const int wave = tid >> 5;
    const int half = lane >> 4;   // 0: lanes 0-15, 1: lanes 16-31
    const int r    = lane & 15;

    __shared__ float Alds[16][16];
    __shared__ float red[8];
    __shared__ float s_scale;

    // ---------------- Phase 1: Sinkhorn (wave 0 only) ----------------
    if (wave == 0) {
        // lane l handles row n = l&15 (upper half duplicates -> no divergence)
        const float* ap = attn + (size_t)b * 256 + (size_t)r * 16;
        float p[16];
        #pragma unroll
        for (int m = 0; m < 16; ++m) p[m] = ap[m] + KEPS;

        for (int it = 0; it < SINKHORN_ITERS; ++it) {
            // row normalization (in-lane across 16 regs)
            float rs = 0.f;
            #pragma unroll
            for (int m = 0; m < 16; ++m) rs += p[m];
            const float rr = __builtin_amdgcn_rcpf(rs);
            #pragma unroll
            for (int m = 0; m < 16; ++m) p[m] *= rr;
            // column normalization (butterfly across the 16-lane group)
            #pragma unroll
            for (int m = 0; m < 16; ++m) {
                float cs = p[m];
                cs += __shfl_xor(cs, 1, 32);
                cs += __shfl_xor(cs, 2, 32);
                cs += __shfl_xor(cs, 4, 32);
                cs += __shfl_xor(cs, 8, 32);
                p[m] *= __builtin_amdgcn_rcpf(cs);
            }
        }
        // Fold identity term: A = 0.9*P + 0.1/256 so that
        // sum_n (A @ S)[n,d] = 0.9*sum_{n,m} P[n,m] S[m,d] + (0.1/16) sum_m S[m,d]
        if (lane < 16) {
            #pragma unroll
            for (int m = 0; m < 16; ++m)
                Alds[r][m] = 0.9f * p[m] + (0.1f / 256.0f);
        }
    }
    __syncthreads();

    // ---------------- Phase 2: WMMA  C = A(16x16) @ S(16x16 chunk) ----------------
    const float* Sb = states + (size_t)b * 16 * 1024;

    // A operand: 4 chunks of 16x4 (M x K), ISA layout:
    //   VGPR0 = {K=4k+2*half}, VGPR1 = {K=4k+2*half+1}, M = lane&15
    v2f a[4];
    #pragma unroll
    for (int k = 0; k < 4; ++k) {
        a[k].x = Alds[r][4 * k + 2 * half + 0];
        a[k].y = Alds[r][4 * k + 2 * half + 1];
    }

    float res[8];
    #pragma unroll
    for (int c = 0; c < 8; ++c) {
        const int dbase = wave * 128 + c * 16;
        v8f acc = {0.f, 0.f, 0.f, 0.f, 0.f, 0.f, 0.f, 0.f};
        #pragma unroll
        for (int k = 0; k < 4; ++k) {
            const int row = 4 * k + 2 * half;   // K index held by this half-wave
            v2f bm;
            bm.x = Sb[(size_t)row * 1024 + dbase + r];
            bm.y = Sb[(size_t)(row + 1) * 1024 + dbase + r];
            // (neg_a, A, neg_b, B, c_mod, C, reuse_a, reuse_b)
            acc = __builtin_amdgcn_wmma_f32_16x16x4_f32(
                false, a[k], false, bm, (short)0, acc, false, false);
        }
        // Reduce C over M: 8 VGPRs cover M=0..7 (half 0) / 8..15 (half 1), then
        // cross-half add. Every lane ends with out[dbase + (lane&15)].
        float t = acc[0] + acc[1] + acc[2] + acc[3] +
                  acc[4] + acc[5] + acc[6] + acc[7];
        t += __shfl_xor(t, 16, 32);
        res[c] = t;
    }

    // ---------------- Norm clamp ----------------
    float ss = 0.f;
    if (half == 0) {                    // avoid double-counting duplicated halves
        #pragma unroll
        for (int c = 0; c < 8; ++c) ss += res[c] * res[c];
    }
    ss += __shfl_xor(ss, 1, 32);
    ss += __shfl_xor(ss, 2, 32);
    ss += __shfl_xor(ss, 4, 32);
    ss += __shfl_xor(ss, 8, 32);
    ss += __shfl_xor(ss, 16, 32);
    if (lane == 0) red[wave] = ss;
    __syncthreads();
    if (tid == 0) {
        float tot = 0.f;
        #pragma unroll
        for (int w = 0; w < 8; ++w) tot += red[w];
        const float nrm = sqrtf(tot);
        const float sc  = 1.0f / (nrm + KEPS);
        s_scale = sc < 1.0f ? sc : 1.0f;
    }
    __syncthreads();

    const float sc = s_scale;
    if (half == 0) {
        float* ob = out + (size_t)b * 1024 + wave * 128 + r;
        #pragma unroll
        for (int c = 0; c < 8; ++c) ob[c * 16] = res[c] * sc;
    }
}

extern "C" void kernel_launch(void* const* d_in, const int* in_sizes, int n_in,
                              void* d_out, int out_size, void* d_ws, size_t ws_size,
                              hipStream_t stream) {
    const float* states = (const float*)d_in[0];   // [B,16,1024]
    const float* attn   = (const float*)d_in[1];   // [B,16,16]
    float* out          = (float*)d_out;           // [B,1024]
    const int B = in_sizes[0] / (16 * 1024);
    mhc_sinkhorn_mix_kernel<<<B, 256, 0, stream>>>(states, attn, out);
}